// MyLoss_68487548502732
// MI455X (gfx1250) — compile-verified
//
#include <hip/hip_runtime.h>
#include <math.h>

typedef float v2f __attribute__((ext_vector_type(2)));
typedef float v8f __attribute__((ext_vector_type(8)));

#define MU 0.1f
#define KGROUPS 64
#define DFEAT 16
// scratch layout (floats): [0..1023] sum[64][16], [1024..2047] sumsq[64][16],
//                          [2048..3071] stats[64][16] (col0=count, col1=sum_i, col2=sum_i2)
#define ACC_FLOATS (3 * KGROUPS * DFEAT)

__device__ __forceinline__ v8f wmma_f32(v2f a, v2f b, v8f c) {
    return __builtin_amdgcn_wmma_f32_16x16x4_f32(false, a, false, b, (short)0, c,
                                                 false, false);
}

// Process one 32-row tile: 8 chunks of K=4 rows, 12 WMMAs per chunk.
// GUARD=false: tile fully in range (fast path, no per-load bounds checks).
template <bool GUARD>
__device__ __forceinline__ void process_tile(
    const float* __restrict__ data, int N, int rowBase, int myLab,
    int hi, int m, bool is_c0, bool is_c1, bool is_c2,
    v8f (&Csum)[4], v8f (&Csq)[4], v8f (&Cst)[4])
{
    // lane covers B rows (rowBase + 4c + 2*hi) and (+1), column m
    const float* pB = data + (size_t)(rowBase + 2 * hi) * DFEAT + m;

    float b0, b1;
    if (GUARD) {
        const int r0 = rowBase + 2 * hi;
        b0 = (r0     < N) ? pB[0]     : 0.0f;
        b1 = (r0 + 1 < N) ? pB[DFEAT] : 0.0f;
    } else {
        b0 = pB[0];
        b1 = pB[DFEAT];
    }

    #pragma unroll 1                       // one compact chunk body
    for (int c = 0; c < 8; ++c) {
        // ---- software-pipelined load of next chunk's B values ----
        float n0, n1;
        if (GUARD) {
            const int r0n = rowBase + (c + 1) * 4 + 2 * hi;
            n0 = (c < 7 && r0n     < N) ? pB[(c + 1) * 64]         : 0.0f;
            n1 = (c < 7 && r0n + 1 < N) ? pB[(c + 1) * 64 + DFEAT] : 0.0f;
        } else {
            const int cn = (c < 7) ? (c + 1) : 7;   // uniform clamp, no OOB
            n0 = pB[cn * 64];
            n1 = pB[cn * 64 + DFEAT];
        }

        const int lab0 = __shfl(myLab, c * 4 + 2 * hi,     32);
        const int lab1 = __shfl(myLab, c * 4 + 2 * hi + 1, 32);

        const v2f B  = {b0, b1};
        const v2f Bq = {b0 * b0, b1 * b1};

        const int r0 = rowBase + c * 4 + 2 * hi;
        const float f0 = (float)r0, f1 = (float)(r0 + 1);
        const float s0 = is_c0 ? 1.0f : is_c1 ? f0 : is_c2 ? f0 * f0 : 0.0f;
        const float s1 = is_c0 ? 1.0f : is_c1 ? f1 : is_c2 ? f1 * f1 : 0.0f;
        const v2f Bs = {s0, s1};

        #pragma unroll
        for (int t = 0; t < 4; ++t) {      // 4 group-tiles of 16 groups, straight-line
            const int gbase = t << 4;
            const v2f A = {(lab0 == gbase + m) ? 1.0f : 0.0f,
                           (lab1 == gbase + m) ? 1.0f : 0.0f};
            Csum[t] = wmma_f32(A, B,  Csum[t]);
            Csq[t]  = wmma_f32(A, Bq, Csq[t]);
            Cst[t]  = wmma_f32(A, Bs, Cst[t]);
        }

        b0 = n0;
        b1 = n1;
    }
}

__global__ __launch_bounds__(256) void seg_accum_kernel(
    const float* __restrict__ data, const int* __restrict__ label,
    float* __restrict__ scratch, int N)
{
    __shared__ float lacc[ACC_FLOATS];
    for (int i = threadIdx.x; i < ACC_FLOATS; i += blockDim.x) lacc[i] = 0.0f;
    __syncthreads();

    const int lane        = threadIdx.x & 31;
    const int waveInBlock = threadIdx.x >> 5;
    const int wavesPerBlk = blockDim.x >> 5;
    const int waveId      = blockIdx.x * wavesPerBlk + waveInBlock;
    const int numWaves    = gridDim.x * wavesPerBlk;

    const int hi = lane >> 4;   // half-wave selector: lane covers K rows 2*hi, 2*hi+1
    const int m  = lane & 15;   // M index for A, feature/N index for B and C

    // column selectors for the stats B matrix (col0 = 1, col1 = idx, col2 = idx^2)
    const bool is_c0 = (m == 0), is_c1 = (m == 1), is_c2 = (m == 2);

    // 12 register accumulator tiles: 4 group-tiles x {sum, sumsq, stats} = 96 VGPRs
    v8f Csum[4] = {}, Csq[4] = {}, Cst[4] = {};

    const int numTiles = (N + 31) >> 5;   // 32 rows per wave-tile
    for (int tile = waveId; tile < numTiles; tile += numWaves) {
        const int rowBase  = tile << 5;
        const bool full    = (rowBase + 32) <= N;   // wave-uniform
        const int  myRow   = rowBase + lane;

        // prefetch this wave's next grid-stride tile (global_prefetch_b8)
        {
            const int nextRow = rowBase + (numWaves << 5) + lane;
            if (nextRow < N)
                __builtin_prefetch(&data[(size_t)nextRow * DFEAT], 0, 1);
        }

        if (full) {
            const int myLab = label[myRow];
            process_tile<false>(data, N, rowBase, myLab, hi, m,
                                is_c0, is_c1, is_c2, Csum, Csq, Cst);
        } else {
            const int myLab = (myRow < N) ? label[myRow] : -1;  // zero one-hot
            process_tile<true>(data, N, rowBase, myLab, hi, m,
                               is_c0, is_c1, is_c2, Csum, Csq, Cst);
        }
    }

    // Drain wave accumulators into block LDS table (ds_add_f32)
    #pragma unroll
    for (int t = 0; t < 4; ++t) {
        #pragma unroll
        for (int j = 0; j < 8; ++j) {
            const int g = (t << 4) + j + (hi << 3);   // C layout: M = j + 8*hi
            atomicAdd(&lacc[0 * 1024 + g * DFEAT + m], Csum[t][j]);
            atomicAdd(&lacc[1 * 1024 + g * DFEAT + m], Csq[t][j]);
            atomicAdd(&lacc[2 * 1024 + g * DFEAT + m], Cst[t][j]);
        }
    }
    __syncthreads();

    // One global atomic pass per block
    for (int i = threadIdx.x; i < ACC_FLOATS; i += blockDim.x)
        atomicAdd(&scratch[i], lacc[i]);
}

__global__ void finalize_kernel(const float* __restrict__ scratch,
                                float* __restrict__ out)
{
    __shared__ float red[KGROUPS];
    const int g = threadIdx.x;

    const float* sums = scratch;
    const float* sq   = scratch + 1024;
    const float* st   = scratch + 2048;

    const float cnt    = st[g * DFEAT + 0];
    const float sum_i  = st[g * DFEAT + 1];
    const float sum_i2 = st[g * DFEAT + 2];
    const float cm     = fmaxf(cnt, 1.0f);

    float sse = 0.0f;
    #pragma unroll
    for (int d = 0; d < DFEAT; ++d) {
        const float mean = sums[g * DFEAT + d] / cm;
        sse += sq[g * DFEAT + d] - cm * mean * mean;
    }
    const float loss2  = sse / cm;

    const float mean_i  = sum_i / cm;
    const float var_i   = (sum_i2 - cm * mean_i * mean_i) / fmaxf(cm - 1.0f, 1.0f);
    const float std_i   = sqrtf(fmaxf(var_i, 0.0f));
    const float min_std = sqrtf(cm * (cm + 1.0f) / 12.0f);
    const float loss1   = (std_i - min_std) / cm;

    const float present = (cnt > 0.0f) ? 1.0f : 0.0f;
    red[g] = present * (MU * loss1 + loss2);
    __syncthreads();

    if (g == 0) {
        float s = 0.0f;
        #pragma unroll
        for (int i = 0; i < KGROUPS; ++i) s += red[i];
        out[0] = s;
    }
}

extern "C" void kernel_launch(void* const* d_in, const int* in_sizes, int n_in,
                              void* d_out, int out_size, void* d_ws, size_t ws_size,
                              hipStream_t stream) {
    (void)n_in; (void)out_size; (void)ws_size;
    const float* data  = (const float*)d_in[0];
    const int*   label = (const int*)d_in[1];
    const int    N     = in_sizes[1];          // 2,000,000 rows (D = 16)
    float* scratch     = (float*)d_ws;

    hipMemsetAsync(scratch, 0, ACC_FLOATS * sizeof(float), stream);

    // 1024 blocks x 8 waves = 8192 waves; each wave owns 12 WMMA accumulator tiles
    seg_accum_kernel<<<1024, 256, 0, stream>>>(data, label, scratch, N);
    finalize_kernel<<<1, KGROUPS, 0, stream>>>(scratch, (float*)d_out);
}